// DifferentialDropout_53506702574028
// MI455X (gfx1250) — compile-verified
//
#include <hip/hip_runtime.h>

typedef __attribute__((ext_vector_type(2))) float v2f;
typedef __attribute__((ext_vector_type(8))) float v8f;

#define BATCH   64
#define DLEN    401408          // 512*28*28
#define NTOT    (64LL * 401408LL)
#define CHUNK   128             // k-slice per LDS buffer
#define KCHUNKS (DLEN / CHUNK)  // 3136
#define NWG_G   256             // gram workgroups (partial buffers)
#define GSTR    (CHUNK + 4)     // 132: padded LDS row stride (floats), conflict-free (132 % 64 == 4)

// ---------------------------------------------------------------- zero presence
__global__ void dd_zero_kernel(unsigned* __restrict__ presence) {
    if (threadIdx.x < 32) presence[threadIdx.x] = 0u;
}

// ---------------------------------------------------------------- Gram partials via V_WMMA_F32_16X16X4_F32
// 128 threads = 4 waves; wave w owns output tile-row w (16x16 tiles).
// Double-buffered LDS panels filled with global_load_async_to_lds_b128 (ASYNCcnt).
__global__ void dd_gram_kernel(const float* __restrict__ x, float* __restrict__ partials) {
    __shared__ float lt[2][64 * GSTR];
    const int tid  = threadIdx.x;
    const int wid  = tid >> 5;          // wave id 0..3
    const int lane = tid & 31;
    const int m    = lane & 15;         // row/col within 16
    const int ks   = (lane >> 4) << 1;  // k sub-offset {0,2}

    v8f acc0 = {}, acc1 = {}, acc2 = {}, acc3 = {};

    // issue async fill of LDS buffer b with k-chunk c (64 rows x CHUNK floats)
    auto issue = [&](int c, int b) {
        const size_t k0 = (size_t)c * CHUNK;
        const int f4pr = CHUNK / 4;                         // 32 float4 per row
        for (int f = tid; f < 64 * f4pr; f += 128) {
            const int row = f / f4pr;
            const int col = (f % f4pr) << 2;
            const unsigned ldsoff =
                (unsigned)(unsigned long long)(const void*)&lt[b][row * GSTR + col];
            const unsigned long long ga =
                (unsigned long long)(const void*)(x + (size_t)row * DLEN + k0 + col);
            asm volatile("global_load_async_to_lds_b128 %0, %1, off"
                         :: "v"(ldsoff), "v"(ga) : "memory");
        }
    };

    int c = blockIdx.x;
    int b = 0;
    if (c < KCHUNKS) issue(c, 0);
    for (; c < KCHUNKS; c += NWG_G) {
        asm volatile("s_wait_asynccnt 0x0" ::: "memory");   // this wave's panel landed
        __syncthreads();                                    // all waves landed + prev compute done
        if (c + NWG_G < KCHUNKS) issue(c + NWG_G, b ^ 1);   // overlap next fill with compute
        const float* __restrict__ panel = &lt[b][0];
        #pragma unroll
        for (int kk = 0; kk < CHUNK; kk += 4) {
            // identical per-lane layout serves as A (row-block) and B (col-block) fragment
            const v2f fa = *(const v2f*)&panel[(wid * 16 + m) * GSTR + kk + ks];
            const v2f b0 = *(const v2f*)&panel[( 0 + m) * GSTR + kk + ks];
            const v2f b1 = *(const v2f*)&panel[(16 + m) * GSTR + kk + ks];
            const v2f b2 = *(const v2f*)&panel[(32 + m) * GSTR + kk + ks];
            const v2f b3 = *(const v2f*)&panel[(48 + m) * GSTR + kk + ks];
            acc0 = __builtin_amdgcn_wmma_f32_16x16x4_f32(false, fa, false, b0, (short)0, acc0, false, false);
            acc1 = __builtin_amdgcn_wmma_f32_16x16x4_f32(false, fa, false, b1, (short)0, acc1, false, false);
            acc2 = __builtin_amdgcn_wmma_f32_16x16x4_f32(false, fa, false, b2, (short)0, acc2, false, false);
            acc3 = __builtin_amdgcn_wmma_f32_16x16x4_f32(false, fa, false, b3, (short)0, acc3, false, false);
        }
        b ^= 1;
    }

    // C/D layout: vgpr v, lane L -> M = v + 8*(L>>4), N = L&15
    float* gout = partials + (size_t)blockIdx.x * 4096;
    const int n  = lane & 15;
    const int mh = (lane >> 4) * 8;
    #pragma unroll
    for (int v = 0; v < 8; ++v) {
        const int mrow = 16 * wid + mh + v;
        gout[mrow * 64 +  0 + n] = acc0[v];
        gout[mrow * 64 + 16 + n] = acc1[v];
        gout[mrow * 64 + 32 + n] = acc2[v];
        gout[mrow * 64 + 48 + n] = acc3[v];
    }
}

// ---------------------------------------------------------------- deterministic fixed-order reduce of partials
__global__ void dd_reduce_kernel(const float* __restrict__ partials, float* __restrict__ G) {
    const int e = blockIdx.x * blockDim.x + threadIdx.x;  // 0..4095
    float sum = 0.0f;
    for (int w = 0; w < NWG_G; ++w) sum += partials[(size_t)w * 4096 + e];
    G[e] = sum;
}

// ---------------------------------------------------------------- per-row sums + unique-value bitmasks
__global__ void dd_rowstats_kernel(const float* __restrict__ x, float* __restrict__ s,
                                   unsigned* __restrict__ rowUnique, unsigned* __restrict__ presence) {
    __shared__ unsigned pres[32];
    __shared__ float red[256];
    const int tid = threadIdx.x;
    const int r   = blockIdx.x;
    if (tid < 32) pres[tid] = 0u;
    __syncthreads();

    const float* xr = x + (size_t)r * DLEN;
    float sum = 0.0f;
    for (int d = tid * 4; d < DLEN; d += 256 * 4) {   // DLEN % 1024 == 0
        const float4 v = *(const float4*)(xr + d);
        const float vv[4] = {v.x, v.y, v.z, v.w};
        #pragma unroll
        for (int q = 0; q < 4; ++q) {
            sum += vv[q];
            const float rv = rintf(vv[q]);           // round-half-even, matches jnp.round
            const int   iv = (int)fminf(fmaxf(rv, -512.0f), 511.0f);
            const unsigned bin = (unsigned)(iv + 512);
            const unsigned w   = bin >> 5;
            const unsigned bit = 1u << (bin & 31);
            if (!(pres[w] & bit)) atomicOr(&pres[w], bit);   // idempotent, deterministic final state
        }
    }
    red[tid] = sum;
    __syncthreads();
    for (int st = 128; st > 0; st >>= 1) {
        if (tid < st) red[tid] += red[tid + st];
        __syncthreads();
    }
    if (tid < 32 && pres[tid]) atomicOr(&presence[tid], pres[tid]);
    if (tid == 0) {
        unsigned cnt = 0;
        for (int i = 0; i < 32; ++i) cnt += __popc(pres[i]);
        rowUnique[r] = cnt;
        s[r] = red[0];
    }
}

// ---------------------------------------------------------------- finalize scalar p from G, s, unique counts
__global__ void dd_finalize_kernel(const float* __restrict__ G, const float* __restrict__ s,
                                   const unsigned* __restrict__ rowUnique,
                                   const unsigned* __restrict__ presence,
                                   float* __restrict__ p_ws, float* __restrict__ out_last) {
    __shared__ float stdv[64], rowg[64], red[64];
    __shared__ float totG, trace;
    const int i = threadIdx.x;                 // 64 threads
    const float Df = (float)DLEN;
    const float invD  = 1.0f / Df;
    const float invD1 = 1.0f / (Df - 1.0f);

    float rg = 0.0f;
    for (int j = 0; j < 64; ++j) rg += G[i * 64 + j];
    rowg[i] = rg;
    const float si = s[i];
    stdv[i] = sqrtf((G[i * 64 + i] - si * si * invD) * invD1);
    __syncthreads();
    if (i == 0) {
        float t = 0.0f, tr = 0.0f;
        for (int j = 0; j < 64; ++j) { t += rowg[j]; tr += G[j * 64 + j]; }
        totG = t; trace = tr;
    }
    __syncthreads();

    float f1 = 0.0f;
    const float inv_si = 1.0f / stdv[i];
    for (int j = 0; j < 64; ++j) {
        const float cov  = (G[i * 64 + j] - si * s[j] * invD) * invD1;
        const float corr = cov * inv_si / stdv[j];
        f1 += fminf(fabsf(corr), 1.0f);        // |clip(corr,-1,1)|
    }
    f1 *= (1.0f / 64.0f);

    const float row_mse   = (G[i * 64 + i] - rowg[i] * (2.0f / 64.0f) + totG * (1.0f / 4096.0f)) * invD;
    const float total_mse = (trace - totG * (1.0f / 64.0f)) * invD;

    unsigned tu = 0;
    for (int w = 0; w < 32; ++w) tu += __popc(presence[w]);
    const float f3 = (float)rowUnique[i] / (float)tu;

    red[i] = (1.0f - f1 + row_mse / total_mse + f3) * (1.0f / 3.0f);
    __syncthreads();
    if (i == 0) {
        float mx = red[0];
        for (int j = 1; j < 64; ++j) mx = fmaxf(mx, red[j]);
        mx = fmaxf(mx, 0.0f);
        *p_ws = mx;
        *out_last = mx;
    }
}

// ---------------------------------------------------------------- streaming dropout (float4)
__global__ void dd_dropout_kernel(const float* __restrict__ x, const float* __restrict__ noise,
                                  const float* __restrict__ p_ws, float* __restrict__ out, long long n4) {
    const float p = *p_ws;
    const float scale = 1.0f / (1.0f - p);
    long long idx = (long long)blockIdx.x * blockDim.x + threadIdx.x;
    const long long stride = (long long)gridDim.x * blockDim.x;
    for (; idx < n4; idx += stride) {
        const float4 xv = ((const float4*)x)[idx];
        const float4 nv = ((const float4*)noise)[idx];
        float4 o;
        o.x = (nv.x >= p) ? xv.x * scale : 0.0f;
        o.y = (nv.y >= p) ? xv.y * scale : 0.0f;
        o.z = (nv.z >= p) ? xv.z * scale : 0.0f;
        o.w = (nv.w >= p) ? xv.w * scale : 0.0f;
        ((float4*)out)[idx] = o;
    }
}

// ----------------------------------------------------------------
extern "C" void kernel_launch(void* const* d_in, const int* in_sizes, int n_in,
                              void* d_out, int out_size, void* d_ws, size_t ws_size,
                              hipStream_t stream) {
    const float* x     = (const float*)d_in[0];
    const float* noise = (const float*)d_in[1];
    float* out = (float*)d_out;

    // workspace layout (~4.2 MB)
    float*    partials  = (float*)d_ws;                       // NWG_G * 4096 floats
    float*    G         = partials + (size_t)NWG_G * 4096;    // 4096
    float*    s         = G + 4096;                           // 64
    unsigned* rowUnique = (unsigned*)(s + 64);                // 64
    unsigned* presence  = rowUnique + 64;                     // 32
    float*    p_ws      = (float*)(presence + 32);            // 1

    dd_zero_kernel<<<1, 32, 0, stream>>>(presence);
    dd_gram_kernel<<<NWG_G, 128, 0, stream>>>(x, partials);
    dd_reduce_kernel<<<16, 256, 0, stream>>>(partials, G);
    dd_rowstats_kernel<<<BATCH, 256, 0, stream>>>(x, s, rowUnique, presence);
    dd_finalize_kernel<<<1, 64, 0, stream>>>(G, s, rowUnique, presence, p_ws, out + NTOT);
    dd_dropout_kernel<<<4096, 256, 0, stream>>>(x, noise, p_ws, out, NTOT / 4);
}